// ProbabilisticMap_37666863186156
// MI455X (gfx1250) — compile-verified
//
#include <hip/hip_runtime.h>
#include <hip/hip_bf16.h>
#include <cstddef>

// ---------------------------------------------------------------------------
// ProbabilisticMap on MI455X (gfx1250)
//   out(b,x,y,t) = exp(-0.5 (p-m)^T S^-1 (p-m)) / (2*pi*sqrt(det S))
// Stage 1: V_WMMA_F32_16X16X4_F32 GEMM  basis(64x8) x cp_data(8x{256|512})
//          -> per-(b,t) mean / covariance tables in d_ws (192 KB, L2-resident)
// Stage 2: pure streaming kernel: 1 thread = fixed (t, y%5), params in
//          registers, 2 FMA + v_exp_f32 + NT store per element.
//          Bound by the 104.9 MB output stream: ~4.5 us at 23.3 TB/s.
// ---------------------------------------------------------------------------

#define MAP_W 64
#define MAP_H 64
#define MAP_T 50
#define NUM_CP 8
#define BATCH 128

typedef __attribute__((ext_vector_type(2))) float v2f;
typedef __attribute__((ext_vector_type(8))) float v8f;

// ws layout (floats): [0, 64*256)       mean matrix M=64(t) x N=256(b*2+c)
//                     [64*256, +64*512) cov  matrix M=64(t) x N=512(b*4+cc)
#define WS_MEAN_OFF 0
#define WS_COV_OFF (64 * 256)
#define WS_FLOATS (64 * 256 + 64 * 512)

#define LOG2E 1.4426950408889634f
#define INV_2PI 0.15915494309189535f

// Branchless Bernstein basis: bs[k] = C(7,k) u^k (1-u)^(7-k), straight-line.
__device__ __forceinline__ void bern7_all(float u, float bs[8]) {
    const float v = 1.0f - u;
    const float u2 = u * u, u3 = u2 * u, u4 = u2 * u2;
    const float u5 = u4 * u, u6 = u4 * u2, u7 = u4 * u3;
    const float v2 = v * v, v3 = v2 * v, v4 = v2 * v2;
    const float v5 = v4 * v, v6 = v4 * v2, v7 = v4 * v3;
    bs[0] = v7;
    bs[1] = 7.0f * u * v6;
    bs[2] = 21.0f * u2 * v5;
    bs[3] = 35.0f * u3 * v4;
    bs[4] = 35.0f * u4 * v3;
    bs[5] = 21.0f * u5 * v2;
    bs[6] = 7.0f * u6 * v;
    bs[7] = u7;
}

// ---------------------------------------------------------------------------
// Stage 1: 192 waves of V_WMMA_F32_16X16X4_F32.
//   waves [0,64):   means (N=256), 4 m-tiles x 16 n-tiles
//   waves [64,192): covs  (N=512, basis^2), 4 m-tiles x 32 n-tiles
// A 16x4 (ISA 7.12.2): lane L -> row M=L%16; VGPR j holds K=(L/16)*2+j.
// B 4x16: lane L -> col N=L%16; VGPR j holds K=(L/16)*2+j.
// C/D 16x16: VGPR r -> row M = r + (L/16)*8, col N = L%16.
// ---------------------------------------------------------------------------
__global__ void pm_stage1_wmma(const float* __restrict__ cp_means,
                               const float* __restrict__ cp_covs,
                               float* __restrict__ ws) {
    const int wave = (blockIdx.x * blockDim.x + threadIdx.x) >> 5;
    const int lane = threadIdx.x & 31;
    const bool isCov = (wave >= 64);          // uniform per wave
    const int wi = isCov ? (wave - 64) : wave;
    const int nTiles = isCov ? 32 : 16;
    const int mt = wi / nTiles;               // 0..3 (t tile)
    const int nt = wi % nTiles;               // column tile
    const float* X = isCov ? cp_covs : cp_means;
    const int stride = isCov ? 512 : 256;
    float* dst = ws + (isCov ? WS_COV_OFF : WS_MEAN_OFF);

    const int half = lane >> 4;    // selects K pair {0,1} vs {2,3} within step
    const int l16 = lane & 15;
    const int row = mt * 16 + l16;            // t index for A
    const int col = nt * 16 + l16;            // N column for B/D
    const float u =
        (float)(row < (MAP_T - 1) ? row : (MAP_T - 1)) / (float)(MAP_T - 1);

    float bs[8];
    bern7_all(u, bs);
    if (isCov) {
#pragma unroll
        for (int i = 0; i < 8; ++i) bs[i] *= bs[i];
    }

    v8f acc = {};
#pragma unroll
    for (int s = 0; s < 2; ++s) {
        const int k0 = s * 4 + half * 2;      // k0 in {0,2} or {4,6}
        v2f A;
        A.x = half ? bs[s * 4 + 2] : bs[s * 4 + 0];
        A.y = half ? bs[s * 4 + 3] : bs[s * 4 + 1];
        v2f B;
        B.x = X[(size_t)k0 * stride + col];
        B.y = X[(size_t)(k0 + 1) * stride + col];
        acc = __builtin_amdgcn_wmma_f32_16x16x4_f32(
            /*neg_a=*/false, A, /*neg_b=*/false, B,
            /*c_mod=*/(short)0, acc, /*reuse_a=*/false, /*reuse_b=*/false);
    }
#pragma unroll
    for (int r = 0; r < 8; ++r) {
        const int m = mt * 16 + half * 8 + r;  // t row of D tile
        dst[(size_t)m * stride + col] = acc[r];
    }
}

// ---------------------------------------------------------------------------
// Stage 2: one block per (b,x); thread tid<250 owns (t=tid%50, ysub=tid/50)
// and walks y += 5. All params loop-invariant in registers; body is
// dy-add + 2 FMA + v_exp_f32 + mul + coalesced NT store.
// Output index: ((b*64 + x)*64 + y)*50 + t.
// ---------------------------------------------------------------------------
__global__ __launch_bounds__(256) void pm_stage2_map(
    const float* __restrict__ ws, float* __restrict__ out) {
    const int tid = threadIdx.x;
    if (tid >= 5 * MAP_T) return;             // 250 active threads
    const int b = blockIdx.x >> 6;
    const int x = blockIdx.x & 63;
    const int t = tid % MAP_T;
    const int ysub = tid / MAP_T;             // 0..4

    const float* wm = ws + WS_MEAN_OFF;       // 64 x 256 (L2-resident)
    const float* wc = ws + WS_COV_OFF;        // 64 x 512
    const float mx = wm[(size_t)t * 256 + b * 2 + 0];
    const float my = wm[(size_t)t * 256 + b * 2 + 1];
    const float ca = wc[(size_t)t * 512 + b * 4 + 0];
    const float cb = wc[(size_t)t * 512 + b * 4 + 1];
    const float cc = wc[(size_t)t * 512 + b * 4 + 2];
    const float cd = wc[(size_t)t * 512 + b * 4 + 3];

    const float det = ca * cd - cb * cc;
    const float idet = 1.0f / det;
    const float qxx = -0.5f * LOG2E * cd * idet;        // exp2-prescaled
    const float qxy = 0.5f * LOG2E * (cb + cc) * idet;
    const float qyy = -0.5f * LOG2E * ca * idet;
    const float amp = INV_2PI * rsqrtf(det);

    const float dx = (float)x - mx;
    const float c0 = qxx * dx * dx;           // loop-invariant
    const float c1 = qxy * dx;

    float dy = (float)ysub - my;
    float* o = out + (size_t)(b * MAP_W + x) * (MAP_H * MAP_T) +
               ysub * MAP_T + t;
#pragma unroll 4
    for (int y = ysub; y < MAP_H; y += 5) {
        const float arg = c0 + dy * (c1 + qyy * dy);    // 2 FMA
        __builtin_nontemporal_store(amp * exp2f(arg), o);
        dy += 5.0f;
        o += 5 * MAP_T;
    }
}

// ---------------------------------------------------------------------------
// Fallback (ws too small): same streaming structure, params computed inline
// from the control points (8-term FMA sum per thread).
// ---------------------------------------------------------------------------
__global__ __launch_bounds__(256) void pm_fused(
    const float* __restrict__ cp_means, const float* __restrict__ cp_covs,
    float* __restrict__ out) {
    const int tid = threadIdx.x;
    if (tid >= 5 * MAP_T) return;
    const int b = blockIdx.x >> 6;
    const int x = blockIdx.x & 63;
    const int t = tid % MAP_T;
    const int ysub = tid / MAP_T;

    float bs[8];
    bern7_all((float)t / (float)(MAP_T - 1), bs);
    float mx = 0.f, my = 0.f, ca = 0.f, cb = 0.f, cc = 0.f, cd = 0.f;
#pragma unroll
    for (int n = 0; n < NUM_CP; ++n) {
        const float bn = bs[n];
        const float b2 = bn * bn;
        mx += bn * cp_means[(size_t)n * (BATCH * 2) + b * 2 + 0];
        my += bn * cp_means[(size_t)n * (BATCH * 2) + b * 2 + 1];
        ca += b2 * cp_covs[(size_t)n * (BATCH * 4) + b * 4 + 0];
        cb += b2 * cp_covs[(size_t)n * (BATCH * 4) + b * 4 + 1];
        cc += b2 * cp_covs[(size_t)n * (BATCH * 4) + b * 4 + 2];
        cd += b2 * cp_covs[(size_t)n * (BATCH * 4) + b * 4 + 3];
    }
    const float det = ca * cd - cb * cc;
    const float idet = 1.0f / det;
    const float qxx = -0.5f * LOG2E * cd * idet;
    const float qxy = 0.5f * LOG2E * (cb + cc) * idet;
    const float qyy = -0.5f * LOG2E * ca * idet;
    const float amp = INV_2PI * rsqrtf(det);

    const float dx = (float)x - mx;
    const float c0 = qxx * dx * dx;
    const float c1 = qxy * dx;

    float dy = (float)ysub - my;
    float* o = out + (size_t)(b * MAP_W + x) * (MAP_H * MAP_T) +
               ysub * MAP_T + t;
#pragma unroll 4
    for (int y = ysub; y < MAP_H; y += 5) {
        const float arg = c0 + dy * (c1 + qyy * dy);
        __builtin_nontemporal_store(amp * exp2f(arg), o);
        dy += 5.0f;
        o += 5 * MAP_T;
    }
}

extern "C" void kernel_launch(void* const* d_in, const int* in_sizes, int n_in,
                              void* d_out, int out_size, void* d_ws,
                              size_t ws_size, hipStream_t stream) {
    const float* cp_means = (const float*)d_in[0];  // (8,128,2)   f32
    const float* cp_covs = (const float*)d_in[1];   // (8,128,2,2) f32
    float* out = (float*)d_out;                     // (128,64,64,50) f32
    (void)in_sizes; (void)n_in; (void)out_size;

    const int mapBlocks = BATCH * MAP_W;  // 8192 blocks

    if (d_ws != nullptr && ws_size >= (size_t)WS_FLOATS * sizeof(float)) {
        float* ws = (float*)d_ws;
        pm_stage1_wmma<<<24, 256, 0, stream>>>(cp_means, cp_covs, ws);
        pm_stage2_map<<<mapBlocks, 256, 0, stream>>>(ws, out);
    } else {
        pm_fused<<<mapBlocks, 256, 0, stream>>>(cp_means, cp_covs, out);
    }
}